// SAGPoolModel_40948218200433
// MI455X (gfx1250) — compile-verified
//
#include <hip/hip_runtime.h>
#include <math.h>

// ---------------- problem constants (match reference) ----------------
#define NB     256
#define NPER   512
#define NN     (NB * NPER)      // 131072 nodes
#define EPER   8192
#define EE     (NB * EPER)      // 2097152 edges
#define HID    128
#define OUTF   10
#define KP1    256
#define KP2    128

typedef __attribute__((ext_vector_type(2))) float v2f;
typedef __attribute__((ext_vector_type(8))) float v8f;

// ---------------- edge degree: deg[dst] += emask (or 1) ----------------
__global__ void k_deg(const int* __restrict__ src, const int* __restrict__ dst,
                      const float* __restrict__ emask, float* __restrict__ deg) {
    int e = blockIdx.x * blockDim.x + threadIdx.x;
    if (e >= EE) return;
    float m = emask ? emask[e] : 1.0f;
    if (m != 0.0f) atomicAdd(&deg[dst[e]], m);
}

// ---------------- feature scatter: agg[dst,:] += x[src,:]*emask ----------------
// one block (128 threads) per edge; channel = threadIdx.x (coalesced)
__global__ void k_scatter_feat(const float* __restrict__ x, const int* __restrict__ src,
                               const int* __restrict__ dst, const float* __restrict__ emask,
                               float* __restrict__ agg) {
    int e = blockIdx.x;
    int c = threadIdx.x;
    float m = emask ? emask[e] : 1.0f;
    if (m == 0.0f) return;
    int s = src[e], d = dst[e];
    atomicAdd(&agg[(size_t)d * HID + c], x[(size_t)s * HID + c] * m);
}

// ---------------- agg /= max(deg,1) ----------------
__global__ void k_div_deg(float* __restrict__ agg, const float* __restrict__ deg) {
    size_t idx = (size_t)blockIdx.x * blockDim.x + threadIdx.x;
    if (idx >= (size_t)NN * HID) return;
    float dv = deg[idx / HID];
    agg[idx] /= fmaxf(dv, 1.0f);
}

// ---------------- fused dual-GEMM + bias + ReLU (+ optional row mask) ----------------
// H[i,n] = relu( sum_k A[i,k]*W1[k,n] + sum_k X[i,k]*W2[k,n] + bias[n] ) * (rowmask?rowmask[i]:1)
// One wave (32 threads) per 16x16 output tile, f32 WMMA 16x16x4, K=128.
__global__ void k_gemm_dual_relu(const float* __restrict__ A,  const float* __restrict__ W1,
                                 const float* __restrict__ X,  const float* __restrict__ W2,
                                 const float* __restrict__ bias, const float* __restrict__ rowmask,
                                 float* __restrict__ H) {
    const int lane = threadIdx.x;           // 0..31
    const int half = lane >> 4;             // 0: K+0/1, 1: K+2/3
    const int l    = lane & 15;
    const int m0   = blockIdx.x * 16;       // node tile
    const int n0   = blockIdx.y * 16;       // output-feature tile

    v8f c = {};
    const float* arow = A + (size_t)(m0 + l) * HID;
    const float* xrow = X + (size_t)(m0 + l) * HID;

    #pragma unroll 8
    for (int k0 = 0; k0 < HID; k0 += 4) {
        v2f a, b;
        a.x = arow[k0 + 2 * half];
        a.y = arow[k0 + 2 * half + 1];
        b.x = W1[(size_t)(k0 + 2 * half)     * HID + n0 + l];
        b.y = W1[(size_t)(k0 + 2 * half + 1) * HID + n0 + l];
        c = __builtin_amdgcn_wmma_f32_16x16x4_f32(false, a, false, b, (short)0, c, false, false);
    }
    #pragma unroll 8
    for (int k0 = 0; k0 < HID; k0 += 4) {
        v2f a, b;
        a.x = xrow[k0 + 2 * half];
        a.y = xrow[k0 + 2 * half + 1];
        b.x = W2[(size_t)(k0 + 2 * half)     * HID + n0 + l];
        b.y = W2[(size_t)(k0 + 2 * half + 1) * HID + n0 + l];
        c = __builtin_amdgcn_wmma_f32_16x16x4_f32(false, a, false, b, (short)0, c, false, false);
    }

    float bv = bias[n0 + l];
    #pragma unroll
    for (int r = 0; r < 8; ++r) {
        int row = m0 + r + 8 * half;
        float v = fmaxf(c[r] + bv, 0.0f);
        if (rowmask) v *= rowmask[row];
        H[(size_t)row * HID + n0 + l] = v;
    }
}

// ---------------- per-node dots: t[i]=h.wrel ; s[i]=h.wroot + b ----------------
__global__ void k_node_dots(const float* __restrict__ h, const float* __restrict__ wrel,
                            const float* __restrict__ wroot, const float* __restrict__ pb,
                            float* __restrict__ t, float* __restrict__ s) {
    int wave = (blockIdx.x * blockDim.x + threadIdx.x) >> 5;
    int lane = threadIdx.x & 31;
    if (wave >= NN) return;
    const float* hr = h + (size_t)wave * HID;
    float a = 0.0f, b = 0.0f;
    #pragma unroll
    for (int k = 0; k < HID; k += 32) {
        float hv = hr[k + lane];
        a += hv * wrel[k + lane];
        b += hv * wroot[k + lane];
    }
    #pragma unroll
    for (int off = 16; off > 0; off >>= 1) {
        a += __shfl_xor(a, off, 32);
        b += __shfl_xor(b, off, 32);
    }
    if (lane == 0) { t[wave] = a; s[wave] = b + pb[0]; }
}

// ---------------- edge score scatter: s[dst] += t[src]*emask ----------------
__global__ void k_edge_score(const float* __restrict__ t, const int* __restrict__ src,
                             const int* __restrict__ dst, const float* __restrict__ emask,
                             float* __restrict__ s) {
    int e = blockIdx.x * blockDim.x + threadIdx.x;
    if (e >= EE) return;
    float m = emask ? emask[e] : 1.0f;
    if (m != 0.0f) atomicAdd(&s[dst[e]], t[src[e]] * m);
}

// ---------------- per-graph top-k by exact rank count (stable tie-break) ----------------
__global__ void k_pool(const float* __restrict__ s, const float* __restrict__ prevMask,
                       int K, float* __restrict__ maskOut) {
    __shared__ float sc[NPER];
    int b = blockIdx.x, j = threadIdx.x;
    int node = b * NPER + j;
    float v = s[node];
    if (prevMask && prevMask[node] == 0.0f) v = -__builtin_inff();
    sc[j] = v;
    __syncthreads();
    int rank = 0;
    for (int jj = 0; jj < NPER; ++jj) {
        float o = sc[jj];
        rank += (o > v) || (o == v && jj < j);
    }
    maskOut[node] = (rank < K) ? 1.0f : 0.0f;
}

// ---------------- gating: h[i,:] *= tanh(s[i]) * mask[i] ----------------
__global__ void k_gate(float* __restrict__ h, const float* __restrict__ s,
                       const float* __restrict__ mask) {
    size_t idx = (size_t)blockIdx.x * blockDim.x + threadIdx.x;
    if (idx >= (size_t)NN * HID) return;
    size_t i = idx / HID;
    h[idx] *= tanhf(s[i]) * mask[i];
}

// ---------------- emask = mask[src]*mask[dst] (fresh write) ----------------
__global__ void k_emask(const float* __restrict__ mask, const int* __restrict__ src,
                        const int* __restrict__ dst, float* __restrict__ em) {
    int e = blockIdx.x * blockDim.x + threadIdx.x;
    if (e >= EE) return;
    em[e] = mask[src[e]] * mask[dst[e]];
}

// ---------------- per-graph mean pool over gated nodes ----------------
__global__ void k_meanpool(const float* __restrict__ h, const float* __restrict__ mask,
                           float* __restrict__ g) {
    __shared__ float cntS;
    int b = blockIdx.x, c = threadIdx.x;  // 128 threads
    if (c == 0) {
        float cnt = 0.0f;
        for (int j = 0; j < NPER; ++j) cnt += mask[b * NPER + j];
        cntS = fmaxf(cnt, 1.0f);
    }
    __syncthreads();
    float acc = 0.0f;
    const float* base = h + (size_t)b * NPER * HID;
    for (int j = 0; j < NPER; ++j) acc += base[(size_t)j * HID + c];
    g[(size_t)b * HID + c] = acc / cntS;
}

// ---------------- final linear 128 -> 10 ----------------
__global__ void k_linear(const float* __restrict__ g, const float* __restrict__ w,
                         const float* __restrict__ bias, float* __restrict__ out) {
    int idx = blockIdx.x * blockDim.x + threadIdx.x;
    if (idx >= NB * OUTF) return;
    int b = idx / OUTF, o = idx % OUTF;
    float acc = bias[o];
    const float* gr = g + (size_t)b * HID;
    for (int k = 0; k < HID; ++k) acc += gr[k] * w[(size_t)k * OUTF + o];
    out[idx] = acc;
}

extern "C" void kernel_launch(void* const* d_in, const int* in_sizes, int n_in,
                              void* d_out, int out_size, void* d_ws, size_t ws_size,
                              hipStream_t stream) {
    // inputs in setup_inputs() dict order
    const float* x      = (const float*)d_in[0];
    const int*   ei     = (const int*)  d_in[1];
    const int*   src    = ei;
    const int*   dst    = ei + EE;
    // d_in[2] = batch (unused: graphs are contiguous NPER blocks)
    const float* c1_wl  = (const float*)d_in[3];
    const float* c1_bl  = (const float*)d_in[4];
    const float* c1_wr  = (const float*)d_in[5];
    const float* p1_wrel= (const float*)d_in[6];
    const float* p1_b   = (const float*)d_in[7];
    const float* p1_wrt = (const float*)d_in[8];
    const float* c2_wl  = (const float*)d_in[9];
    const float* c2_bl  = (const float*)d_in[10];
    const float* c2_wr  = (const float*)d_in[11];
    const float* p2_wrel= (const float*)d_in[12];
    const float* p2_b   = (const float*)d_in[13];
    const float* p2_wrt = (const float*)d_in[14];
    const float* lin_w  = (const float*)d_in[15];
    const float* lin_b  = (const float*)d_in[16];
    float* out = (float*)d_out;

    // workspace carve-up
    float* ws = (float*)d_ws;
    size_t o = 0;
    float* h1   = ws + o; o += (size_t)NN * HID;
    float* h2   = ws + o; o += (size_t)NN * HID;
    float* agg  = ws + o; o += (size_t)NN * HID;
    float* tbuf = ws + o; o += NN;
    float* sbuf = ws + o; o += NN;
    float* deg  = ws + o; o += NN;
    float* m1   = ws + o; o += NN;
    float* m2   = ws + o; o += NN;
    float* em   = ws + o; o += EE;
    float* g    = ws + o; o += (size_t)NB * HID;

    const int TB = 256;
    dim3 gemmGrid(NN / 16, HID / 16);
    size_t featElems = (size_t)NN * HID;

    // ======== SAGEConv 1 (all edges active) ========
    hipMemsetAsync(agg, 0, featElems * sizeof(float), stream);
    hipMemsetAsync(deg, 0, (size_t)NN * sizeof(float), stream);
    k_deg<<<(EE + TB - 1) / TB, TB, 0, stream>>>(src, dst, nullptr, deg);
    k_scatter_feat<<<EE, HID, 0, stream>>>(x, src, dst, nullptr, agg);
    k_div_deg<<<(unsigned)((featElems + TB - 1) / TB), TB, 0, stream>>>(agg, deg);
    k_gemm_dual_relu<<<gemmGrid, 32, 0, stream>>>(agg, c1_wl, x, c1_wr, c1_bl, nullptr, h1);

    // ======== score 1 + pool 1 ========
    k_node_dots<<<NN / 8, TB, 0, stream>>>(h1, p1_wrel, p1_wrt, p1_b, tbuf, sbuf);
    k_edge_score<<<(EE + TB - 1) / TB, TB, 0, stream>>>(tbuf, src, dst, nullptr, sbuf);
    k_pool<<<NB, NPER, 0, stream>>>(sbuf, nullptr, KP1, m1);
    k_gate<<<(unsigned)((featElems + TB - 1) / TB), TB, 0, stream>>>(h1, sbuf, m1);
    k_emask<<<(EE + TB - 1) / TB, TB, 0, stream>>>(m1, src, dst, em);

    // ======== SAGEConv 2 (masked edges) ========
    hipMemsetAsync(agg, 0, featElems * sizeof(float), stream);
    hipMemsetAsync(deg, 0, (size_t)NN * sizeof(float), stream);
    k_deg<<<(EE + TB - 1) / TB, TB, 0, stream>>>(src, dst, em, deg);
    k_scatter_feat<<<EE, HID, 0, stream>>>(h1, src, dst, em, agg);
    k_div_deg<<<(unsigned)((featElems + TB - 1) / TB), TB, 0, stream>>>(agg, deg);
    k_gemm_dual_relu<<<gemmGrid, 32, 0, stream>>>(agg, c2_wl, h1, c2_wr, c2_bl, m1, h2);

    // ======== score 2 + pool 2 ========
    k_node_dots<<<NN / 8, TB, 0, stream>>>(h2, p2_wrel, p2_wrt, p2_b, tbuf, sbuf);
    k_edge_score<<<(EE + TB - 1) / TB, TB, 0, stream>>>(tbuf, src, dst, em, sbuf);
    k_pool<<<NB, NPER, 0, stream>>>(sbuf, m1, KP2, m2);
    k_gate<<<(unsigned)((featElems + TB - 1) / TB), TB, 0, stream>>>(h2, sbuf, m2);

    // ======== readout ========
    k_meanpool<<<NB, HID, 0, stream>>>(h2, m2, g);
    k_linear<<<(NB * OUTF + TB - 1) / TB, TB, 0, stream>>>(g, lin_w, lin_b, out);
}